// BiparGAT_55825984913937
// MI455X (gfx1250) — compile-verified
//
#include <hip/hip_runtime.h>
#include <math.h>

// ---------------- problem constants (from reference) ----------------
#define HID 128
#define NH  4
#define DH  32
#define NL  2
#define NB  16
#define NV  100000
#define NC  50000
#define NE  500000
#define DV  16
#define DC  8
#define NFV 4
#define NFC 8

typedef float v8f __attribute__((ext_vector_type(8)));
typedef float v2f __attribute__((ext_vector_type(2)));

// ---------------- small device helpers ----------------
__device__ __forceinline__ void atomAddF(float* p, float v) {
  __hip_atomic_fetch_add(p, v, __ATOMIC_RELAXED, __HIP_MEMORY_SCOPE_AGENT);
}
// order-preserving float<->uint key for atomicMax-based segment max
__device__ __forceinline__ unsigned fkey(float f) {
  unsigned u = __float_as_uint(f);
  return (u & 0x80000000u) ? ~u : (u | 0x80000000u);
}
__device__ __forceinline__ float fval(unsigned k) {
  unsigned u = (k & 0x80000000u) ? (k ^ 0x80000000u) : ~k;
  return __uint_as_float(u);
}

// ---------------- fill / init kernels ----------------
__global__ void k_fill_f32(float* p, float v, int n) {
  int i = blockIdx.x * 256 + threadIdx.x;
  if (i < n) p[i] = v;
}
__global__ void k_fill_u32(unsigned* p, unsigned v, int n) {
  int i = blockIdx.x * 256 + threadIdx.x;
  if (i < n) p[i] = v;
}
__global__ void k_init_bias(float* out, const float* __restrict__ bias, int total) {
  int i = blockIdx.x * 256 + threadIdx.x;
  if (i < total) out[i] = bias[i & (HID - 1)];
}
__global__ void k_count(const int* __restrict__ batch, float* cnt, int n) {
  int i = blockIdx.x * 256 + threadIdx.x;
  if (i < n) atomAddF(&cnt[batch[i]], 1.0f);
}

// ---------------- Fourier features ----------------
// col = id*(2*nf) + r ; r<nf -> sin(x*2^r*pi), else cos(x*2^(r-nf)*pi)
__global__ void k_fourier(const float* __restrict__ x, float* __restrict__ out,
                          int n, int din, int nf) {
  int idx = blockIdx.x * 256 + threadIdx.x;
  if (idx >= n * HID) return;
  int node = idx >> 7;
  int c = idx & (HID - 1);
  int two_nf = 2 * nf;
  int id = c / two_nf;
  int r = c - id * two_nf;
  int iscos = (r >= nf);
  int j = iscos ? (r - nf) : r;
  float f = (float)(1 << j) * 3.14159265358979323846f;
  float v = x[node * din + id] * f;
  out[idx] = iscos ? __cosf(v) : __sinf(v);
}

// ---------------- WMMA fp32 GEMM: out[N,128] = in[N,128] @ W[128,128] + b ----
// Block: 256 threads (8 waves), 32 rows per block.
// Wave w: row-tile (w>>2)*16, col-base (w&3)*32 -> two 16x16 accumulators.
// W is staged ONCE per block into LDS, transposed (Wt[col][k]) via one
// coalesced 64KB float4 sweep, so every B fragment is a single aligned
// ds_load_b64 (K-pairs contiguous). MI455X WGP has 320KB LDS; this block
// uses ~83KB (A tile 16.9KB + Wt 66.6KB), leaving multi-block residency.
// Fragment layouts per CDNA5 ISA 7.12.2:
//   A 16x4:  lane m=lane%16, VGPR{0,1} hold K = k0+2*half + {0,1}
//   B 4x16:  lane n=lane%16, VGPR{0,1} hold rows K = k0+2*half + {0,1}
//   C 16x16: VGPR v, lanes 0-15 -> M=v, lanes 16-31 -> M=v+8, N=lane%16
__global__ __launch_bounds__(256) void k_gemm128(
    const float* __restrict__ A, const float* __restrict__ W,
    const float* __restrict__ bias, float* __restrict__ Out, int n) {
  __shared__ float As[32][132];   // A tile, stride keeps b64 pairs aligned
  __shared__ float Wt[HID][130];  // transposed weights: Wt[c][k] = W[k][c]
  const int rowBase = blockIdx.x * 32;
  const int tid = threadIdx.x;
  // cooperative A-tile load (32x128 floats as float4, coalesced)
  for (int i = tid; i < 32 * 32; i += 256) {
    int r = i >> 5, c4 = i & 31;
    int row = rowBase + r;
    float4 v = make_float4(0.f, 0.f, 0.f, 0.f);
    if (row < n) v = ((const float4*)(A + (size_t)row * HID))[c4];
    As[r][c4 * 4 + 0] = v.x;
    As[r][c4 * 4 + 1] = v.y;
    As[r][c4 * 4 + 2] = v.z;
    As[r][c4 * 4 + 3] = v.w;
  }
  // cooperative W stage + transpose (128x128 floats as float4, coalesced)
  for (int i = tid; i < HID * 32; i += 256) {
    int k = i >> 5, c4 = i & 31;
    float4 v = ((const float4*)(W + (size_t)k * HID))[c4];
    int c = c4 * 4;
    Wt[c + 0][k] = v.x;
    Wt[c + 1][k] = v.y;
    Wt[c + 2][k] = v.z;
    Wt[c + 3][k] = v.w;
  }
  __syncthreads();
  const int wave = tid >> 5;
  const int lane = tid & 31;
  const int half = lane >> 4;
  const int ln = lane & 15;
  const int rtile = (wave >> 2) * 16;
  const int cbase = (wave & 3) * 32;
  v8f acc0 = {0, 0, 0, 0, 0, 0, 0, 0};
  v8f acc1 = {0, 0, 0, 0, 0, 0, 0, 0};
  const int m = rtile + ln;
#pragma unroll
  for (int k0 = 0; k0 < HID; k0 += 4) {
    const int kr = k0 + 2 * half;
    v2f a;
    a.x = As[m][kr];
    a.y = As[m][kr + 1];
    v2f b0, b1;
    b0.x = Wt[cbase + ln][kr];
    b0.y = Wt[cbase + ln][kr + 1];
    b1.x = Wt[cbase + 16 + ln][kr];
    b1.y = Wt[cbase + 16 + ln][kr + 1];
    acc0 = __builtin_amdgcn_wmma_f32_16x16x4_f32(false, a, false, b0,
                                                 (short)0, acc0, false, false);
    acc1 = __builtin_amdgcn_wmma_f32_16x16x4_f32(false, a, false, b1,
                                                 (short)0, acc1, false, false);
  }
#pragma unroll
  for (int v = 0; v < 8; ++v) {
    int row = rowBase + rtile + v + 8 * half;
    if (row < n) {
      int c0 = cbase + ln;
      Out[(size_t)row * HID + c0] = acc0[v] + bias[c0];
      Out[(size_t)row * HID + c0 + 16] = acc1[v] + bias[c0 + 16];
    }
  }
}

// ---------------- GATv2 edge kernels ----------------
// wave per edge; lane covers feature f = h*32 + lane for each head h.
__global__ __launch_bounds__(256) void k_edge_score(
    const int* __restrict__ src, const int* __restrict__ dst,
    const float* __restrict__ ea, const float* __restrict__ XL,
    const float* __restrict__ XR, const float* __restrict__ We,
    const float* __restrict__ att, float* __restrict__ escore,
    unsigned* __restrict__ mx, int ne) {
  int e = blockIdx.x * 8 + (threadIdx.x >> 5);
  if (e >= ne) return;
  int lane = threadIdx.x & 31;
  int s = src[e], d = dst[e];
  float eav = ea[e];
  float p[NH];
#pragma unroll
  for (int h = 0; h < NH; ++h) {
    int f = h * 32 + lane;
    float m = XL[(size_t)s * HID + f] + XR[(size_t)d * HID + f] + eav * We[f];
    m = (m > 0.f) ? m : 0.2f * m;  // leaky_relu(0.2)
    p[h] = m * att[f];
  }
#pragma unroll
  for (int off = 16; off > 0; off >>= 1) {
#pragma unroll
    for (int h = 0; h < NH; ++h) p[h] += __shfl_xor(p[h], off, 32);
  }
  if (lane < NH) {
    escore[(size_t)e * NH + lane] = p[lane];
    atomicMax(&mx[(size_t)d * NH + lane], fkey(p[lane]));
  }
}

__global__ void k_edge_exp(const int* __restrict__ dst,
                           float* __restrict__ escore,
                           const unsigned* __restrict__ mx,
                           float* __restrict__ sm, int ne) {
  int idx = blockIdx.x * 256 + threadIdx.x;
  if (idx >= ne * NH) return;
  int e = idx >> 2, h = idx & 3;
  int d = dst[e];
  float m = fval(mx[(size_t)d * NH + h]);
  float ex = __expf(escore[idx] - m);
  escore[idx] = ex;
  atomAddF(&sm[(size_t)d * NH + h], ex);
}

__global__ __launch_bounds__(256) void k_edge_aggr(
    const int* __restrict__ src, const int* __restrict__ dst,
    const float* __restrict__ escore, const float* __restrict__ sm,
    const float* __restrict__ XL, float* __restrict__ nout, int ne) {
  int e = blockIdx.x * 8 + (threadIdx.x >> 5);
  if (e >= ne) return;
  int lane = threadIdx.x & 31;
  int s = src[e], d = dst[e];
#pragma unroll
  for (int h = 0; h < NH; ++h) {
    float a = escore[(size_t)e * NH + h] / (sm[(size_t)d * NH + h] + 1e-16f);
    int f = h * 32 + lane;
    atomAddF(&nout[(size_t)d * HID + f], a * XL[(size_t)s * HID + f]);
  }
}

// ---------------- residual + GraphNorm ----------------
__global__ void k_res_lrelu(float* __restrict__ nout, const float* __restrict__ x,
                            int total) {
  int i = blockIdx.x * 256 + threadIdx.x;
  if (i < total) {
    float v = nout[i] + x[i];
    nout[i] = (v > 0.f) ? v : 0.01f * v;
  }
}
__global__ void k_bsum(const float* __restrict__ x, const int* __restrict__ batch,
                       float* bsum, int n) {
  int idx = blockIdx.x * 256 + threadIdx.x;
  if (idx >= n * HID) return;
  int node = idx >> 7, f = idx & (HID - 1);
  atomAddF(&bsum[batch[node] * HID + f], x[idx]);
}
__global__ void k_bscale(float* s, const float* __restrict__ cnt, int n) {
  int i = blockIdx.x * 256 + threadIdx.x;
  if (i < n) {
    float c = cnt[i >> 7];
    s[i] /= (c > 0.f ? c : 1.f);
  }
}
__global__ void k_center_var(float* __restrict__ h, const int* __restrict__ batch,
                             const float* __restrict__ mean,
                             const float* __restrict__ ms, float* bvar, int n) {
  int idx = blockIdx.x * 256 + threadIdx.x;
  if (idx >= n * HID) return;
  int node = idx >> 7, f = idx & (HID - 1);
  float xc = h[idx] - ms[f] * mean[batch[node] * HID + f];
  h[idx] = xc;
  atomAddF(&bvar[batch[node] * HID + f], xc * xc);
}
__global__ void k_norm_write(const float* __restrict__ h,
                             const int* __restrict__ batch,
                             const float* __restrict__ var_,
                             const float* __restrict__ w,
                             const float* __restrict__ b,
                             float* __restrict__ xout, int n) {
  int idx = blockIdx.x * 256 + threadIdx.x;
  if (idx >= n * HID) return;
  int node = idx >> 7, f = idx & (HID - 1);
  xout[idx] = w[f] * h[idx] * rsqrtf(var_[batch[node] * HID + f] + 1e-5f) + b[f];
}

// ---------------- readout head: [16,256] MLPs in one block ----------------
__global__ __launch_bounds__(256) void k_head(
    const float* __restrict__ gv, const float* __restrict__ gc,
    const float* __restrict__ W1, const float* __restrict__ b1,
    const float* __restrict__ lw1, const float* __restrict__ lb1,
    const float* __restrict__ W2, const float* __restrict__ b2,
    const float* __restrict__ W3, const float* __restrict__ b3,
    const float* __restrict__ lw2, const float* __restrict__ lb2,
    const float* __restrict__ W4, const float* __restrict__ b4,
    float* __restrict__ out) {
  __shared__ float xg[NB][256];
  __shared__ float h1[NB][256];
  __shared__ float mu[NB], rsd[NB];
  int t = threadIdx.x;
  for (int i = t; i < NB * 256; i += 256) {
    int b = i >> 8, f = i & 255;
    xg[b][f] = (f < HID) ? gv[b * HID + f] : gc[b * HID + (f - HID)];
  }
  __syncthreads();
  {  // h1 = xg @ W1 + b1   (W1: [256][256])
    float col[NB];
#pragma unroll
    for (int b = 0; b < NB; ++b) col[b] = b1[t];
    for (int k = 0; k < 256; ++k) {
      float w = W1[k * 256 + t];
#pragma unroll
      for (int b = 0; b < NB; ++b) col[b] += xg[b][k] * w;
    }
#pragma unroll
    for (int b = 0; b < NB; ++b) h1[b][t] = col[b];
  }
  __syncthreads();
  if (t < NB) {  // row LayerNorm stats
    float m = 0.f;
    for (int k = 0; k < 256; ++k) m += h1[t][k];
    m *= (1.f / 256.f);
    float v = 0.f;
    for (int k = 0; k < 256; ++k) { float d = h1[t][k] - m; v += d * d; }
    v *= (1.f / 256.f);
    mu[t] = m;
    rsd[t] = rsqrtf(v + 1e-5f);
  }
  __syncthreads();
#pragma unroll
  for (int b = 0; b < NB; ++b) {
    float y = (h1[b][t] - mu[b]) * rsd[b] * lw1[t] + lb1[t];
    h1[b][t] = (y > 0.f) ? y : 0.f;
  }
  __syncthreads();
  if (t < HID) {  // pe = h1 @ W2 [256,128] + b2 -> xg[:,0:128]
    float col[NB];
#pragma unroll
    for (int b = 0; b < NB; ++b) col[b] = b2[t];
    for (int k = 0; k < 256; ++k) {
      float w = W2[k * HID + t];
#pragma unroll
      for (int b = 0; b < NB; ++b) col[b] += h1[b][k] * w;
    }
#pragma unroll
    for (int b = 0; b < NB; ++b) xg[b][t] = col[b];
  }
  __syncthreads();
  {  // h2 = pe @ W3 [128,256] + b3 -> h1
    float col[NB];
#pragma unroll
    for (int b = 0; b < NB; ++b) col[b] = b3[t];
    for (int k = 0; k < HID; ++k) {
      float w = W3[k * 256 + t];
#pragma unroll
      for (int b = 0; b < NB; ++b) col[b] += xg[b][k] * w;
    }
#pragma unroll
    for (int b = 0; b < NB; ++b) h1[b][t] = col[b];
  }
  __syncthreads();
  if (t < NB) {
    float m = 0.f;
    for (int k = 0; k < 256; ++k) m += h1[t][k];
    m *= (1.f / 256.f);
    float v = 0.f;
    for (int k = 0; k < 256; ++k) { float d = h1[t][k] - m; v += d * d; }
    v *= (1.f / 256.f);
    mu[t] = m;
    rsd[t] = rsqrtf(v + 1e-5f);
  }
  __syncthreads();
#pragma unroll
  for (int b = 0; b < NB; ++b) {
    float y = (h1[b][t] - mu[b]) * rsd[b] * lw2[t] + lb2[t];
    h1[b][t] = (y > 0.f) ? y : 0.f;
  }
  __syncthreads();
  if (t < NB) {  // out = h2 @ W4 [256,1] + b4
    float acc = b4[0];
    for (int k = 0; k < 256; ++k) acc += h1[t][k] * W4[k];
    out[t] = acc;
  }
}

// ---------------- host orchestration ----------------
#define G256(n) ((unsigned)(((n) + 255) / 256))

extern "C" void kernel_launch(void* const* d_in, const int* in_sizes, int n_in,
                              void* d_out, int out_size, void* d_ws, size_t ws_size,
                              hipStream_t stream) {
  (void)in_sizes; (void)n_in; (void)out_size; (void)ws_size;
  const float* x_vars = (const float*)d_in[0];
  const float* x_cons = (const float*)d_in[1];
  const int* src_cv = (const int*)d_in[2];
  const int* dst_cv = (const int*)d_in[3];
  const float* ea_cv = (const float*)d_in[4];
  const int* src_vc = (const int*)d_in[5];
  const int* dst_vc = (const int*)d_in[6];
  const float* ea_vc = (const float*)d_in[7];
  const int* batch_vars = (const int*)d_in[8];
  const int* batch_cons = (const int*)d_in[9];
  const float* var_W = (const float*)d_in[10];
  const float* var_b = (const float*)d_in[11];
  const float* cons_W = (const float*)d_in[12];
  const float* cons_b = (const float*)d_in[13];
  const float* Wl = (const float*)d_in[14];
  const float* bl = (const float*)d_in[15];
  const float* Wr = (const float*)d_in[16];
  const float* br = (const float*)d_in[17];
  const float* att = (const float*)d_in[18];
  const float* We = (const float*)d_in[19];
  const float* gbias = (const float*)d_in[20];
  const float* gn_w = (const float*)d_in[21];
  const float* gn_b = (const float*)d_in[22];
  const float* gn_ms = (const float*)d_in[23];
  const float* pe_W1 = (const float*)d_in[24];
  const float* pe_b1 = (const float*)d_in[25];
  const float* pe_lw = (const float*)d_in[26];
  const float* pe_lb = (const float*)d_in[27];
  const float* pe_W2 = (const float*)d_in[28];
  const float* pe_b2 = (const float*)d_in[29];
  const float* pm_W1 = (const float*)d_in[30];
  const float* pm_b1 = (const float*)d_in[31];
  const float* pm_lw = (const float*)d_in[32];
  const float* pm_lb = (const float*)d_in[33];
  const float* pm_W2 = (const float*)d_in[34];
  const float* pm_b2 = (const float*)d_in[35];
  float* out = (float*)d_out;

  // workspace carve-up (floats)
  float* ws = (float*)d_ws;
  size_t o = 0;
  auto carve = [&](size_t nflt) { float* p = ws + o; o += nflt; return p; };
  float* xv    = carve((size_t)NV * HID);
  float* xc    = carve((size_t)NC * HID);
  float* xl_cv = carve((size_t)NC * HID);  // also reused as Fc before layers
  float* xr_cv = carve((size_t)NV * HID);
  float* xl_vc = carve((size_t)NV * HID);  // also reused as Fv before layers
  float* xr_vc = carve((size_t)NC * HID);
  float* nv    = carve((size_t)NV * HID);
  float* nc    = carve((size_t)NC * HID);
  float* e_cv  = carve((size_t)NE * NH);
  float* e_vc  = carve((size_t)NE * NH);
  unsigned* mx_v = (unsigned*)carve((size_t)NV * NH);
  float* sm_v  = carve((size_t)NV * NH);
  unsigned* mx_c = (unsigned*)carve((size_t)NC * NH);
  float* sm_c  = carve((size_t)NC * NH);
  float* cnt_v = carve(NB);
  float* cnt_c = carve(NB);
  float* bstat = carve(NB * HID);
  float* bstat2 = carve(NB * HID);

  // --- embeddings: fourier -> WMMA GEMM ---
  float* Fv = xl_vc;  // [NV,128], free until layer transforms
  float* Fc = xl_cv;  // [NC,128]
  k_fourier<<<G256(NV * HID), 256, 0, stream>>>(x_vars, Fv, NV, DV, NFV);
  k_fourier<<<G256(NC * HID), 256, 0, stream>>>(x_cons, Fc, NC, DC, NFC);
  k_gemm128<<<(NV + 31) / 32, 256, 0, stream>>>(Fv, var_W, var_b, xv, NV);
  k_gemm128<<<(NC + 31) / 32, 256, 0, stream>>>(Fc, cons_W, cons_b, xc, NC);

  // --- per-batch node counts (reused every GraphNorm + readout) ---
  k_fill_f32<<<1, 256, 0, stream>>>(cnt_v, 0.f, NB);
  k_fill_f32<<<1, 256, 0, stream>>>(cnt_c, 0.f, NB);
  k_count<<<G256(NV), 256, 0, stream>>>(batch_vars, cnt_v, NV);
  k_count<<<G256(NC), 256, 0, stream>>>(batch_cons, cnt_c, NC);

  auto graphnorm = [&](float* nout, const int* batch, float* cnt, float* xio, int n) {
    k_fill_f32<<<G256(NB * HID), 256, 0, stream>>>(bstat, 0.f, NB * HID);
    k_bsum<<<G256(n * HID), 256, 0, stream>>>(nout, batch, bstat, n);
    k_bscale<<<G256(NB * HID), 256, 0, stream>>>(bstat, cnt, NB * HID);
    k_fill_f32<<<G256(NB * HID), 256, 0, stream>>>(bstat2, 0.f, NB * HID);
    k_center_var<<<G256(n * HID), 256, 0, stream>>>(nout, batch, bstat, gn_ms, bstat2, n);
    k_bscale<<<G256(NB * HID), 256, 0, stream>>>(bstat2, cnt, NB * HID);
    k_norm_write<<<G256(n * HID), 256, 0, stream>>>(nout, batch, bstat2, gn_w, gn_b, xio, n);
  };

  const unsigned egrid = (NE + 7) / 8;
  for (int l = 0; l < NL; ++l) {
    const size_t wo0 = (size_t)(0 * NL + l) * HID * HID;  // rel 0: cons->vars
    const size_t wo1 = (size_t)(1 * NL + l) * HID * HID;  // rel 1: vars->cons
    const size_t vo0 = (size_t)(0 * NL + l) * HID;
    const size_t vo1 = (size_t)(1 * NL + l) * HID;
    // node transforms (WMMA GEMMs) from pre-layer xv/xc
    k_gemm128<<<(NC + 31) / 32, 256, 0, stream>>>(xc, Wl + wo0, bl + vo0, xl_cv, NC);
    k_gemm128<<<(NV + 31) / 32, 256, 0, stream>>>(xv, Wr + wo0, br + vo0, xr_cv, NV);
    k_gemm128<<<(NV + 31) / 32, 256, 0, stream>>>(xv, Wl + wo1, bl + vo1, xl_vc, NV);
    k_gemm128<<<(NC + 31) / 32, 256, 0, stream>>>(xc, Wr + wo1, br + vo1, xr_vc, NC);
    // relation 0: cons -> vars
    k_fill_u32<<<G256(NV * NH), 256, 0, stream>>>(mx_v, 0u, NV * NH);
    k_fill_f32<<<G256(NV * NH), 256, 0, stream>>>(sm_v, 0.f, NV * NH);
    k_edge_score<<<egrid, 256, 0, stream>>>(src_cv, dst_cv, ea_cv, xl_cv, xr_cv,
                                            We + vo0, att + vo0, e_cv, mx_v, NE);
    k_edge_exp<<<G256(NE * NH), 256, 0, stream>>>(dst_cv, e_cv, mx_v, sm_v, NE);
    k_init_bias<<<G256(NV * HID), 256, 0, stream>>>(nv, gbias + vo0, NV * HID);
    k_edge_aggr<<<egrid, 256, 0, stream>>>(src_cv, dst_cv, e_cv, sm_v, xl_cv, nv, NE);
    // relation 1: vars -> cons
    k_fill_u32<<<G256(NC * NH), 256, 0, stream>>>(mx_c, 0u, NC * NH);
    k_fill_f32<<<G256(NC * NH), 256, 0, stream>>>(sm_c, 0.f, NC * NH);
    k_edge_score<<<egrid, 256, 0, stream>>>(src_vc, dst_vc, ea_vc, xl_vc, xr_vc,
                                            We + vo1, att + vo1, e_vc, mx_c, NE);
    k_edge_exp<<<G256(NE * NH), 256, 0, stream>>>(dst_vc, e_vc, mx_c, sm_c, NE);
    k_init_bias<<<G256(NC * HID), 256, 0, stream>>>(nc, gbias + vo1, NC * HID);
    k_edge_aggr<<<egrid, 256, 0, stream>>>(src_vc, dst_vc, e_vc, sm_c, xl_vc, nc, NE);
    // residual + leaky relu (both use pre-update xv/xc), then GraphNorm -> xv/xc
    k_res_lrelu<<<G256(NV * HID), 256, 0, stream>>>(nv, xv, NV * HID);
    k_res_lrelu<<<G256(NC * HID), 256, 0, stream>>>(nc, xc, NC * HID);
    graphnorm(nv, batch_vars, cnt_v, xv, NV);
    graphnorm(nc, batch_cons, cnt_c, xc, NC);
  }

  // --- readout: segment means -> head MLPs ---
  k_fill_f32<<<G256(NB * HID), 256, 0, stream>>>(bstat, 0.f, NB * HID);
  k_fill_f32<<<G256(NB * HID), 256, 0, stream>>>(bstat2, 0.f, NB * HID);
  k_bsum<<<G256(NV * HID), 256, 0, stream>>>(xv, batch_vars, bstat, NV);
  k_bsum<<<G256(NC * HID), 256, 0, stream>>>(xc, batch_cons, bstat2, NC);
  k_bscale<<<G256(NB * HID), 256, 0, stream>>>(bstat, cnt_v, NB * HID);
  k_bscale<<<G256(NB * HID), 256, 0, stream>>>(bstat2, cnt_c, NB * HID);
  k_head<<<1, 256, 0, stream>>>(bstat, bstat2, pe_W1, pe_b1, pe_lw, pe_lb,
                                pe_W2, pe_b2, pm_W1, pm_b1, pm_lw, pm_lb,
                                pm_W2, pm_b2, out);
}